// Detector_90881507983396
// MI455X (gfx1250) — compile-verified
//
#include <hip/hip_runtime.h>
#include <stdint.h>

#define NB 16
#define WW 512
#define HW (512*512)
#define TOPK 2000
#define MAXOUT 500
#define CAND 4096
#define BINS 8192
#define NCHUNK (TOPK/16)
#define KEPT_PAD 2016
#define NWAVES 8

typedef __attribute__((ext_vector_type(2))) float v2f;
typedef __attribute__((ext_vector_type(8))) float v8f;

// workspace layout (bytes)
#define WS_THRESH 0                         // u32[16]
#define WS_COUNT  64                        // u32[16]
#define WS_KEYS   128                       // u64[16*4096]
#define WS_SC     (WS_KEYS + NB*CAND*8)     // f32[16*2000]
#define WS_LY     (WS_SC + NB*TOPK*4)
#define WS_LX     (WS_LY + NB*TOPK*4)

__global__ void k_init(uint8_t* ws) {
    unsigned t = threadIdx.x;
    if (t < NB) ((unsigned*)(ws + WS_COUNT))[t] = 0u;
}

__global__ __launch_bounds__(256) void k_hist(const float* __restrict__ scores, uint8_t* ws) {
    __shared__ unsigned hist[BINS];                 // 32 KB LDS
    const int b = blockIdx.x, tid = threadIdx.x;
    for (int i = tid; i < BINS; i += 256) hist[i] = 0u;
    __syncthreads();
    const float* s = scores + (size_t)b * HW;
    for (int i = tid; i < HW; i += 256) {
        int bin = (int)(s[i] * (float)BINS);
        bin = bin < 0 ? 0 : (bin > BINS - 1 ? BINS - 1 : bin);
        atomicAdd(&hist[bin], 1u);
    }
    __syncthreads();
    if (tid == 0) {
        unsigned acc = 0; int t = 0;
        for (int bin = BINS - 1; bin >= 0; --bin) {
            acc += hist[bin];
            if (acc >= TOPK) { t = bin; break; }
        }
        ((unsigned*)(ws + WS_THRESH))[b] = (unsigned)t;
    }
}

__global__ __launch_bounds__(256) void k_gather(const float* __restrict__ scores, uint8_t* ws) {
    const int b = blockIdx.x, tid = threadIdx.x;
    const unsigned T = ((const unsigned*)(ws + WS_THRESH))[b];
    unsigned* cnt = ((unsigned*)(ws + WS_COUNT)) + b;
    unsigned long long* keys = ((unsigned long long*)(ws + WS_KEYS)) + (size_t)b * CAND;
    const float* s = scores + (size_t)b * HW;
    for (int i = tid; i < HW; i += 256) {
        float v = s[i];
        int bin = (int)(v * (float)BINS);
        bin = bin < 0 ? 0 : (bin > BINS - 1 ? BINS - 1 : bin);
        if ((unsigned)bin >= T) {
            unsigned pos = atomicAdd(cnt, 1u);
            if (pos < CAND)
                keys[pos] = ((unsigned long long)__float_as_uint(v) << 32) |
                            (unsigned long long)(~(unsigned)i);
        }
    }
}

__global__ __launch_bounds__(1024) void k_sort(const float* __restrict__ regs, uint8_t* ws) {
    __shared__ unsigned long long sk[CAND];         // 32 KB LDS
    const int b = blockIdx.x, tid = threadIdx.x;
    unsigned cnt = ((const unsigned*)(ws + WS_COUNT))[b];
    if (cnt > CAND) cnt = CAND;
    const unsigned long long* keys = ((const unsigned long long*)(ws + WS_KEYS)) + (size_t)b * CAND;
    for (int i = tid; i < CAND; i += 1024) sk[i] = (i < (int)cnt) ? keys[i] : 0ull;
    // bitonic sort, descending (keys are {score_bits, ~idx} -> top_k tie order)
    for (unsigned k = 2; k <= CAND; k <<= 1) {
        for (unsigned j = k >> 1; j > 0; j >>= 1) {
            __syncthreads();
            for (unsigned i = tid; i < CAND; i += 1024) {
                unsigned ixj = i ^ j;
                if (ixj > i) {
                    unsigned long long va = sk[i], vb = sk[ixj];
                    bool asc = (i & k) != 0;
                    if ((va < vb) != asc) { sk[i] = vb; sk[ixj] = va; }
                }
            }
        }
    }
    __syncthreads();
    float* sc = (float*)(ws + WS_SC) + (size_t)b * TOPK;
    float* ly = (float*)(ws + WS_LY) + (size_t)b * TOPK;
    float* lx = (float*)(ws + WS_LX) + (size_t)b * TOPK;
    for (int i = tid; i < TOPK; i += 1024) {
        unsigned long long key = sk[i];
        float score; unsigned idx;
        if (key == 0ull) { score = 0.0f; idx = 0u; }
        else { score = __uint_as_float((unsigned)(key >> 32)); idx = ~((unsigned)key); }
        float y = (float)(idx >> 9);
        float x = (float)(idx & (WW - 1));
        const float* r = regs + ((size_t)b * HW + idx) * 2;
        sc[i] = score;
        ly[i] = y + 0.5f + r[0];
        lx[i] = x + 0.5f + r[1];
    }
}

__global__ __launch_bounds__(256) void k_nms(const uint8_t* __restrict__ ws, float* __restrict__ out) {
    __shared__ float ky[KEPT_PAD], kx[KEPT_PAD], ks[KEPT_PAD];   // kept pts (centered coords)
    __shared__ float cy[16], cx[16], cs[16], csc[16];
    __shared__ float d2buf[256];
    __shared__ unsigned supmask;
    __shared__ int keptCount;

    const int b = blockIdx.x, tid = threadIdx.x;
    const int wave = tid >> 5, lane = tid & 31;
    const float* sc = (const float*)(ws + WS_SC) + (size_t)b * TOPK;
    const float* ly = (const float*)(ws + WS_LY) + (size_t)b * TOPK;
    const float* lx = (const float*)(ws + WS_LX) + (size_t)b * TOPK;
    float* out_loc = out;                    // [NB, MAXOUT, 2]
    float* out_sc  = out + NB * MAXOUT * 2;  // [NB, MAXOUT]

    for (int i = tid; i < KEPT_PAD; i += 256) { ky[i] = 1e9f; kx[i] = 1e9f; ks[i] = 1e9f; }
    for (int i = tid; i < MAXOUT; i += 256) {
        out_sc[b * MAXOUT + i] = -1.0f;
        out_loc[(b * MAXOUT + i) * 2 + 0] = -1.0f;
        out_loc[(b * MAXOUT + i) * 2 + 1] = -1.0f;
    }
    if (tid == 0) keptCount = 0;
    __syncthreads();

    for (int c = 0; c < NCHUNK; ++c) {
        const int base = c * 16;
        if (tid < 16) {
            float yy = ly[base + tid] - 256.0f;   // center to limit f32 cancellation
            float xx = lx[base + tid] - 256.0f;
            csc[tid] = sc[base + tid];
            cy[tid] = yy; cx[tid] = xx; cs[tid] = yy * yy + xx * xx;
        }
        if (tid == 0) supmask = 0u;
        __syncthreads();

        const int K = keptCount;
        const int nTiles = (K + 15) >> 4;

        // A[i] = [y_i, x_i, s_i, 1]  (16x4 f32, wave32 layout: lanes 0-15 hold K0/K1, 16-31 hold K2/K3)
        v2f a;
        if (lane < 16) { a.x = cy[lane];        a.y = cx[lane]; }
        else           { a.x = cs[lane - 16];   a.y = 1.0f;     }

        // 16 new proposals vs all previously-kept, 16x16 d2 tiles via V_WMMA_F32_16X16X4_F32
        for (int t = wave; t < nTiles; t += NWAVES) {
            const int n = (t << 4) + (lane & 15);
            // B[.][j] = [-2y_j, -2x_j, 1, s_j]  -> A.B = d2(i,j)
            v2f bb;
            if (lane < 16) { bb.x = -2.0f * ky[n]; bb.y = -2.0f * kx[n]; }
            else           { bb.x = 1.0f;          bb.y = ks[n];         }
            v8f cacc = {0.f, 0.f, 0.f, 0.f, 0.f, 0.f, 0.f, 0.f};
            v8f d = __builtin_amdgcn_wmma_f32_16x16x4_f32(
                false, a, false, bb, (short)0, cacc, false, false);
            unsigned lm = 0;
            const int mbase = (lane >= 16) ? 8 : 0;
#pragma unroll
            for (int v = 0; v < 8; ++v)
                if (d[v] < 1.0f) lm |= (1u << (mbase + v));   // thresh^2 = 1.0
            if (lm) atomicOr(&supmask, lm);
        }

        // intra-chunk 16x16 distances (256 pairs, one per thread)
        {
            const int i = tid >> 4, j = tid & 15;
            float dy = cy[i] - cy[j], dx = cx[i] - cx[j];
            d2buf[tid] = dy * dy + dx * dx;
        }
        __syncthreads();

        // exact sequential greedy within the chunk
        if (tid == 0) {
            const unsigned sup = supmask;
            unsigned keptIn = 0;
            int Kc = keptCount;
            for (int i = 0; i < 16; ++i) {
                bool s = (sup >> i) & 1u;
                if (!s) {
                    for (int j = 0; j < i; ++j)
                        if (((keptIn >> j) & 1u) && d2buf[i * 16 + j] < 1.0f) { s = true; break; }
                }
                if (!s && csc[i] >= 0.5f) {
                    keptIn |= 1u << i;
                    if (Kc < KEPT_PAD) { ky[Kc] = cy[i]; kx[Kc] = cx[i]; ks[Kc] = cs[i]; }
                    if (Kc < MAXOUT) {
                        out_sc[b * MAXOUT + Kc] = csc[i];
                        out_loc[(b * MAXOUT + Kc) * 2 + 0] = ly[base + i];
                        out_loc[(b * MAXOUT + Kc) * 2 + 1] = lx[base + i];
                    }
                    Kc++;
                }
            }
            keptCount = Kc;
        }
        __syncthreads();
    }
}

extern "C" void kernel_launch(void* const* d_in, const int* in_sizes, int n_in,
                              void* d_out, int out_size, void* d_ws, size_t ws_size,
                              hipStream_t stream) {
    const float* scores = (const float*)d_in[0];   // [16,512,512,1]
    const float* regs   = (const float*)d_in[1];   // [16,512,512,2]
    (void)d_in; (void)in_sizes; (void)n_in; (void)out_size; (void)ws_size;
    uint8_t* ws = (uint8_t*)d_ws;
    float* out = (float*)d_out;

    k_init  <<<1, 64, 0, stream>>>(ws);
    k_hist  <<<NB, 256, 0, stream>>>(scores, ws);
    k_gather<<<NB, 256, 0, stream>>>(scores, ws);
    k_sort  <<<NB, 1024, 0, stream>>>(regs, ws);
    k_nms   <<<NB, 256, 0, stream>>>(ws, out);
}